// SNNNet_30923764531666
// MI455X (gfx1250) — compile-verified
//
#include <hip/hip_runtime.h>
#include <stdint.h>

typedef float v2f __attribute__((ext_vector_type(2)));
typedef float v8f __attribute__((ext_vector_type(8)));

#define M_ROWS 16384
#define N_COLS 512
#define K_DIM  784
#define BM 128
#define BN 64
#define KB 16
#define LSTR 20   // LDS row stride (dwords): 16 distinct banks for 16-lane row-strided
                  // fragment reads; 80B rows keep 16B alignment for b128 async stores

#define XBUF (BM * LSTR)   // one x slab   (10 KB / buffer)
#define WBUF (BN * LSTR)   // one W1 slab  ( 5 KB / buffer)

#define BETA 0.9f
#define THR  1.0f

// Per-lane async copy of 16 bytes global->LDS (GLOBAL_LOAD_ASYNC_TO_LDS_B128,
// tracked by ASYNCcnt). Inline asm = portable across both toolchains.
__device__ __forceinline__ void async_g2l_b128(uint32_t lds_off, uint64_t gaddr) {
  asm volatile("global_load_async_to_lds_b128 %0, %1, off"
               :: "v"(lds_off), "v"(gaddr) : "memory");
}
// ASYNCcnt watermark waits: async loads complete in order, so <=3 means
// "everything except the 3 most recently issued is done".
__device__ __forceinline__ void wait_async_le3() {
  asm volatile("s_wait_asynccnt 3" ::: "memory");
}
__device__ __forceinline__ void wait_async_le0() {
  asm volatile("s_wait_asynccnt 0" ::: "memory");
}

// ---------------------------------------------------------------------------
// Kernel 1: cur1[M,512] = x[M,784] @ W1[512,784]^T + b1   (fp32 WMMA 16x16x4)
// 256 thr = 8 waves; 128x64 C tile; DOUBLE-BUFFERED async-LDS K-slab pipeline.
// ---------------------------------------------------------------------------
__global__ __launch_bounds__(256) void snn_gemm1_wmma(
    const float* __restrict__ x,
    const float* __restrict__ W1,
    const float* __restrict__ b1,
    float* __restrict__ cur1)
{
  __shared__ alignas(16) float xs[2 * XBUF];   // x tiles  [2][128 rows][16 k]
  __shared__ alignas(16) float wsh[2 * WBUF];  // W1 tiles [2][64 rows][16 k]

  const int tid  = threadIdx.x;
  const int wave = tid >> 5;
  const int lane = tid & 31;
  const int l16  = lane & 15;
  const int hi   = lane >> 4;

  const int gm0 = blockIdx.y * BM;
  const int gn0 = blockIdx.x * BN;

  v8f acc[4] = {};   // wave strip: 16 rows x 64 cols = 4 WMMA C tiles

  const int lr = tid >> 2;          // 0..63: staging row
  const int lc = (tid & 3) << 2;    // 0,4,8,12: staging col (float4)

  const uint32_t xs_o0 = (uint32_t)(size_t)(xs  + lr * LSTR + lc);
  const uint32_t xs_o1 = (uint32_t)(size_t)(xs  + (lr + 64) * LSTR + lc);
  const uint32_t ws_o  = (uint32_t)(size_t)(wsh + lr * LSTR + lc);
  const float* xrow0 = x  + (size_t)(gm0 + lr)      * K_DIM + lc;
  const float* xrow1 = x  + (size_t)(gm0 + lr + 64) * K_DIM + lc;
  const float* wrow  = W1 + (size_t)(gn0 + lr)      * K_DIM + lc;

  // prologue: stage slab 0 into buffer 0
  async_g2l_b128(xs_o0, (uint64_t)(size_t)(xrow0));
  async_g2l_b128(xs_o1, (uint64_t)(size_t)(xrow1));
  async_g2l_b128(ws_o,  (uint64_t)(size_t)(wrow));

  int buf = 0;
  for (int kb = 0; kb < K_DIM; kb += KB) {
    const bool more = (kb + KB) < K_DIM;
    if (more) {  // stage NEXT slab into the other buffer; stays in flight
      const uint32_t nb = (uint32_t)(buf ^ 1);
      async_g2l_b128(xs_o0 + nb * (XBUF * 4), (uint64_t)(size_t)(xrow0 + kb + KB));
      async_g2l_b128(xs_o1 + nb * (XBUF * 4), (uint64_t)(size_t)(xrow1 + kb + KB));
      async_g2l_b128(ws_o  + nb * (WBUF * 4), (uint64_t)(size_t)(wrow  + kb + KB));
      wait_async_le3();    // current slab's 3 writes landed; next slab's in flight
    } else {
      wait_async_le0();
    }
    __syncthreads();       // publish current slab to all waves

    const float* arow = xs + buf * XBUF + (wave * 16 + l16) * LSTR;
    const float* bbase = wsh + buf * WBUF;
    #pragma unroll
    for (int ks = 0; ks < KB; ks += 4) {
      const int kk = ks + 2 * hi;             // ISA A layout: v0=K, v1=K+1; hi lanes K+2
      const v2f a = *(const v2f*)(arow + kk);
      #pragma unroll
      for (int t = 0; t < 4; ++t) {
        const v2f b = *(const v2f*)(bbase + (t * 16 + l16) * LSTR + kk);
        acc[t] = __builtin_amdgcn_wmma_f32_16x16x4_f32(
            false, a, false, b, (short)0, acc[t], false, false);
      }
    }
    __syncthreads();       // all waves done reading this buffer before it is reused
    buf ^= 1;
  }

  // C layout: VGPR r -> (M=r, N=l16) lanes 0-15, (M=r+8) lanes 16-31
  const int orow = gm0 + wave * 16 + hi * 8;
  #pragma unroll
  for (int t = 0; t < 4; ++t) {
    const int col = gn0 + t * 16 + l16;
    const float bb = b1[col];
    #pragma unroll
    for (int r = 0; r < 8; ++r)
      cur1[(size_t)(orow + r) * N_COLS + col] = acc[t][r] + bb;
  }
}

// ---------------------------------------------------------------------------
// Kernel 2: fused LIF step loop, spk1@W2^T on the WMMA pipe.
// Block = 16 batch rows, 8 waves; wave w owns K-chunk [64w, 64w+64).
// mem1/cur1 registers ARE the f32 16x16x4 A-fragments (lane l16 = row,
// reg pair = k,k+1; hi lanes k+2,k+3), so spikes feed WMMA with zero LDS.
// B-fragments (W2^T, cols 10..15 zeroed) hoisted to 32 regs, loaded once.
// ---------------------------------------------------------------------------
__global__ __launch_bounds__(256) void snn_steps_wmma(
    const float* __restrict__ cur1,   // [M, 512]
    const float* __restrict__ W2,     // [10, 512]
    const float* __restrict__ b2,     // [10]
    const int*   __restrict__ nsteps,
    float* __restrict__ out)          // [steps, M, 10]
{
  __shared__ float cred[8 * 256];     // per-wave partial C: [wave][m*16 + n]

  const int tid  = threadIdx.x;
  const int wave = tid >> 5;
  const int lane = tid & 31;
  const int l16  = lane & 15;
  const int hi   = lane >> 4;

  const int row0  = blockIdx.x * 16;
  const int kbase = wave * 64;

  // B fragments: b[ks] = W2t[kbase+4ks+2hi .. +1][l16], zero for cols >= 10
  v2f bfrag[16];
  #pragma unroll
  for (int ks = 0; ks < 16; ++ks) {
    const int k = kbase + ks * 4 + 2 * hi;
    if (l16 < 10) {
      bfrag[ks].x = W2[l16 * N_COLS + k];
      bfrag[ks].y = W2[l16 * N_COLS + k + 1];
    } else {
      bfrag[ks].x = 0.0f; bfrag[ks].y = 0.0f;
    }
  }

  // cur1 fragment + mem1 state in A-fragment layout
  float cur[32], mem[32];
  const float* crow = cur1 + (size_t)(row0 + l16) * N_COLS + kbase + 2 * hi;
  #pragma unroll
  for (int ks = 0; ks < 16; ++ks) {
    const v2f c = *(const v2f*)(crow + ks * 4);
    cur[2 * ks]     = c.x;  cur[2 * ks + 1]  = c.y;
    mem[2 * ks]     = 0.0f; mem[2 * ks + 1]  = 0.0f;
  }

  // layer-2 state: threads 0..159 own one (row, output) pair
  float mem2 = 0.0f;
  const int orow = tid / 10, oo = tid % 10;
  const float bb2 = (tid < 160) ? b2[oo] : 0.0f;
  const int steps = *nsteps;

  for (int s = 0; s < steps; ++s) {
    v8f acc = {};
    #pragma unroll
    for (int ks = 0; ks < 16; ++ks) {
      v2f a;
      #pragma unroll
      for (int h = 0; h < 2; ++h) {
        float m = mem[2 * ks + h];
        const float rst = (m > THR) ? 1.0f : 0.0f;   // reset from previous-step mem
        m = BETA * m + cur[2 * ks + h] - rst * THR;
        mem[2 * ks + h] = m;
        const float sp = (m - THR > 0.0f) ? 1.0f : 0.0f;
        if (h == 0) a.x = sp; else a.y = sp;
      }
      acc = __builtin_amdgcn_wmma_f32_16x16x4_f32(
          false, a, false, bfrag[ks], (short)0, acc, false, false);
    }

    // cross-wave partial-C reduction (8 KB LDS, ~4K dwords/step)
    #pragma unroll
    for (int r = 0; r < 8; ++r)
      cred[wave * 256 + (r + 8 * hi) * 16 + l16] = acc[r];
    __syncthreads();

    if (tid < 160) {
      float c2 = bb2;
      #pragma unroll
      for (int w = 0; w < 8; ++w) c2 += cred[w * 256 + orow * 16 + oo];
      const float r2 = (mem2 > THR) ? 1.0f : 0.0f;
      mem2 = BETA * mem2 + c2 - r2 * THR;
      out[(size_t)s * (M_ROWS * 10) + (size_t)(row0 + orow) * 10 + oo] =
          (mem2 - THR > 0.0f) ? 1.0f : 0.0f;
    }
    __syncthreads();
  }
}

// ---------------------------------------------------------------------------
extern "C" void kernel_launch(void* const* d_in, const int* in_sizes, int n_in,
                              void* d_out, int out_size, void* d_ws, size_t ws_size,
                              hipStream_t stream) {
  const float* x   = (const float*)d_in[0];   // [16384, 784]
  const float* W1  = (const float*)d_in[1];   // [512, 784]
  const float* b1  = (const float*)d_in[2];   // [512]
  const float* W2  = (const float*)d_in[3];   // [10, 512]
  const float* b2  = (const float*)d_in[4];   // [10]
  const int* steps = (const int*)d_in[5];     // scalar
  float* out  = (float*)d_out;
  float* cur1 = (float*)d_ws;                 // 16384*512*4 = 32 MiB scratch

  dim3 gridA(N_COLS / BN, M_ROWS / BM);       // (8, 128)
  snn_gemm1_wmma<<<gridA, 256, 0, stream>>>(x, W1, b1, cur1);
  snn_steps_wmma<<<M_ROWS / 16, 256, 0, stream>>>(cur1, W2, b2, steps, out);
}